// POEM_Projective_SelfAggregation_Head_66099546685926
// MI455X (gfx1250) — compile-verified
//
#include <hip/hip_runtime.h>

// ---------------------------------------------------------------------------
// POEM projective self-aggregation head, gather-then-compute schedule.
// B=4 N=8 C=256 H=W=32 E=256 D=64 K=32 P=1024. fp32 throughout, using
// V_WMMA_F32_16X16X4_F32 (CDNA5 fp32 matrix op) for every GEMM-shaped stage.
// ---------------------------------------------------------------------------

typedef float v2f __attribute__((ext_vector_type(2)));
typedef float v8f __attribute__((ext_vector_type(8)));

static __device__ __forceinline__ v8f wmma4(v2f a, v2f b, v8f c) {
  // D(16x16,f32) = A(16x4,f32) x B(4x16,f32) + C
  return __builtin_amdgcn_wmma_f32_16x16x4_f32(
      false, a, false, b, (short)0, c, false, false);
}

// One 16x16 output tile, A in LDS (row-major, lda), B in global (row-major, ldb),
// accumulating over Kdim in steps of 4.
// ISA 7.12.2 layouts: A frag: lanes0-15 row=lane, v0=K0,v1=K1; lanes16-31 v0=K2,v1=K3.
// B frag (4x16): v0 = rows K0 (lanes0-15) / K2 (lanes16-31), v1 = K1 / K3.
static __device__ __forceinline__ v8f gemm_tile(const float* __restrict__ A, int lda,
                                                const float* __restrict__ Bg, int ldb,
                                                int n0, int Kdim, int lane, v8f acc) {
  const int m    = lane & 15;
  const int koff = (lane >> 4) << 1;           // 0 or 2
  const float* arow = A + m * lda;
  const float* bcol = Bg + n0 + m;
  for (int kk = 0; kk < Kdim; kk += 4) {
    v2f a; a.x = arow[kk + koff];              a.y = arow[kk + koff + 1];
    v2f b; b.x = bcol[(kk + koff) * ldb];      b.y = bcol[(kk + koff + 1) * ldb];
    acc = wmma4(a, b, acc);
  }
  return acc;
}

// ---------------------------------------------------------------------------
// Kernel 1: ball query (first-K within radius), camera 0, depth bin D/2.
// ---------------------------------------------------------------------------
__global__ __launch_bounds__(256) void k_ball(const float* __restrict__ cam_intr,
                                              const float* __restrict__ cam_extr,
                                              const float* __restrict__ center,
                                              int* __restrict__ idx_out,
                                              float* __restrict__ mask_out) {
  __shared__ float sd2[1024];
  const int b = blockIdx.x;
  const float* In = cam_intr + b * 8 * 9;      // camera 0
  const float* Ex = cam_extr + b * 8 * 16;
  const float fx = In[0], cx = In[2], fy = In[4], cy = In[5];
  const float cpx = center[b * 3 + 0], cpy = center[b * 3 + 1], cpz = center[b * 3 + 2];
  const float dd = 0.1f + (1.4f / 4160.0f) * 32.0f * 33.0f;   // coords_d[32]

  for (int p = threadIdx.x; p < 1024; p += 256) {
    const int hh = p >> 5, ww = p & 31;
    const float x = (ww * 8.0f - cx) / fx * dd;
    const float y = (hh * 8.0f - cy) / fy * dd;
    const float ax = Ex[0] * x + Ex[1] * y + Ex[2]  * dd + Ex[3];
    const float ay = Ex[4] * x + Ex[5] * y + Ex[6]  * dd + Ex[7];
    const float az = Ex[8] * x + Ex[9] * y + Ex[10] * dd + Ex[11];
    const float dx = ax - cpx, dy = ay - cpy, dz = az - cpz;
    sd2[p] = dx * dx + dy * dy + dz * dz;
  }
  __syncthreads();
  if (threadIdx.x == 0) {
    int cnt = 0;
    for (int p = 0; p < 1024 && cnt < 32; ++p)
      if (sd2[p] < 4.0f) { idx_out[b * 32 + cnt] = p; mask_out[b * 32 + cnt] = 1.0f; ++cnt; }
    for (; cnt < 32; ++cnt) { idx_out[b * 32 + cnt] = 0; mask_out[b * 32 + cnt] = 0.0f; }
  }
}

// ---------------------------------------------------------------------------
// Kernel 2: gathered q = (posembed MLP + img projection + biases) * mask.
// 64 blocks x 16 rows; rows r = ((b*32+k)*8+n).
// ---------------------------------------------------------------------------
__global__ __launch_bounds__(256) void k_qproj(const float* __restrict__ img_feat,
                                               const float* __restrict__ cam_intr,
                                               const float* __restrict__ cam_extr,
                                               const float* __restrict__ pe_w1,
                                               const float* __restrict__ pe_b1,
                                               const float* __restrict__ pe_w2,
                                               const float* __restrict__ pe_b2,
                                               const float* __restrict__ proj_w,
                                               const float* __restrict__ proj_b,
                                               const int* __restrict__ idxb,
                                               const float* __restrict__ maskb,
                                               float* __restrict__ qout) {
  __shared__ float sX1[16 * 192];   // inverse-sigmoid features
  __shared__ float sH1[16 * 512];   // hidden layer of PE MLP
  __shared__ float sIMG[16 * 256];  // gathered image channels
  __shared__ float sMask[16];

  const int tid = threadIdx.x;
  const int r0  = blockIdx.x * 16;

  // warm L2 for streamed weights (global_prefetch_b8)
  __builtin_prefetch(pe_w1 + tid * 64, 0, 1);
  __builtin_prefetch(pe_w2 + tid * 64, 0, 1);
  __builtin_prefetch(proj_w + tid * 64, 0, 1);

  if (tid < 16) {
    const int r = r0 + tid;
    sMask[tid] = maskb[(r >> 8) * 32 + ((r >> 3) & 31)];
  }

  // Phase 1: per-row 192-dim geometric feature (64 depths x 3 coords).
  const float bin = 1.4f / 4160.0f;
  for (int t = tid; t < 16 * 64; t += 256) {
    const int i = t >> 6, di = t & 63;
    const int r = r0 + i;
    const int b = r >> 8, k = (r >> 3) & 31, n = r & 7;
    const int p = idxb[b * 32 + k];
    const int hh = p >> 5, ww = p & 31;
    const float* In = cam_intr + (b * 8 + n) * 9;
    const float* Ex = cam_extr + (b * 8 + n) * 16;
    const float fx = In[0], cx = In[2], fy = In[4], cy = In[5];
    const float dd = 0.1f + bin * (float)di * (float)(di + 1);
    const float x = (ww * 8.0f - cx) / fx * dd;
    const float y = (hh * 8.0f - cy) / fy * dd;
    const float lo[3]  = {-1.2f, -1.2f, -0.2f};
    const float isp[3] = {1.0f / 2.4f, 1.0f / 2.4f, 1.0f / 2.0f};
#pragma unroll
    for (int j = 0; j < 3; ++j) {
      const float aj = Ex[j*4+0] * x + Ex[j*4+1] * y + Ex[j*4+2] * dd + Ex[j*4+3];
      float v = (aj - lo[j]) * isp[j];
      v = fminf(fmaxf(v, 0.0f), 1.0f);
      const float x1 = fmaxf(v, 1e-5f), x2 = fmaxf(1.0f - v, 1e-5f);
      sX1[i * 192 + di * 3 + j] = __logf(x1 / x2);
    }
  }

  // Phase 1b: gather image channels (strided 4KB channel gather, 1 MB total).
  for (int t = tid; t < 16 * 256; t += 256) {
    const int i = t >> 8, cc = t & 255;
    const int r = r0 + i;
    const int b = r >> 8, k = (r >> 3) & 31, n = r & 7;
    const int p = idxb[b * 32 + k];
    sIMG[i * 256 + cc] = img_feat[(((b * 8 + n) * 256 + cc) * 32 + (p >> 5)) * 32 + (p & 31)];
  }
  __syncthreads();

  const int lane = tid & 31, wv = tid >> 5;
  const int m = lane & 15, mb = (lane >> 4) * 8;

  // Phase 2: H1 = relu(X1 @ pe_w1 + pe_b1)  (16x512, 32 N-tiles over 8 waves)
  for (int nt = wv; nt < 32; nt += 8) {
    const int n0 = nt * 16;
    v8f acc = {};
    acc = gemm_tile(sX1, 192, pe_w1, 512, n0, 192, lane, acc);
    const float bv = pe_b1[n0 + m];
#pragma unroll
    for (int ii = 0; ii < 8; ++ii) {
      const float v = acc[ii] + bv;
      sH1[(mb + ii) * 512 + n0 + m] = v > 0.0f ? v : 0.0f;
    }
  }
  __syncthreads();

  // Phase 3: q = (H1 @ pe_w2 + IMG @ proj_w + biases) * mask  (16x256)
  for (int nt = wv; nt < 16; nt += 8) {
    const int n0 = nt * 16;
    v8f acc = {};
    acc = gemm_tile(sH1, 512, pe_w2, 256, n0, 512, lane, acc);
    acc = gemm_tile(sIMG, 256, proj_w, 256, n0, 256, lane, acc);
    const float bv = pe_b2[n0 + m] + proj_b[n0 + m];
#pragma unroll
    for (int ii = 0; ii < 8; ++ii) {
      const int i = mb + ii;
      qout[(r0 + i) * 256 + n0 + m] = (acc[ii] + bv) * sMask[i];
    }
  }
}

// ---------------------------------------------------------------------------
// Kernel 3: per-(b,k) head: h = MLP(q), dot-attention aggregate, merge, regress.
// One block per (b,k); N=8 rows zero-padded to a 16-row WMMA tile.
// ---------------------------------------------------------------------------
__global__ __launch_bounds__(256) void k_head(const float* __restrict__ qbuf,
                                              const float* __restrict__ m0a_w,
                                              const float* __restrict__ m0a_b,
                                              const float* __restrict__ m0b_w,
                                              const float* __restrict__ m0b_b,
                                              const float* __restrict__ m1a_w,
                                              const float* __restrict__ m1a_b,
                                              const float* __restrict__ m1b_w,
                                              const float* __restrict__ m1b_b,
                                              const float* __restrict__ reg_w1,
                                              const float* __restrict__ reg_b1,
                                              const float* __restrict__ reg_w2,
                                              const float* __restrict__ reg_b2,
                                              float* __restrict__ out) {
  __shared__ float sQ[16 * 256];
  __shared__ float sH0[16 * 256];
  __shared__ float sH[16 * 128];
  __shared__ float sA[8];
  __shared__ float sAgg[128];
  __shared__ float sT1[128];
  __shared__ float sQm[256];
  __shared__ float sT2[256];

  const int tid = threadIdx.x;
  const int bk  = blockIdx.x;

  for (int t = tid; t < 16 * 256; t += 256) {
    const int i = t >> 8;
    sQ[t] = (i < 8) ? qbuf[(bk * 8 + i) * 256 + (t & 255)] : 0.0f;
  }
  __syncthreads();

  const int lane = tid & 31, wv = tid >> 5;
  const int m = lane & 15, mb = (lane >> 4) * 8;

  // H0 = relu(Q @ m0a_w + b)   (16x256)
  for (int nt = wv; nt < 16; nt += 8) {
    const int n0 = nt * 16;
    v8f acc = {};
    acc = gemm_tile(sQ, 256, m0a_w, 256, n0, 256, lane, acc);
    const float bv = m0a_b[n0 + m];
#pragma unroll
    for (int ii = 0; ii < 8; ++ii) {
      const float v = acc[ii] + bv;
      sH0[(mb + ii) * 256 + n0 + m] = v > 0.0f ? v : 0.0f;
    }
  }
  __syncthreads();

  // H = H0 @ m0b_w + b   (16x128)
  for (int nt = wv; nt < 8; nt += 8) {
    const int n0 = nt * 16;
    v8f acc = {};
    acc = gemm_tile(sH0, 256, m0b_w, 128, n0, 256, lane, acc);
    const float bv = m0b_b[n0 + m];
#pragma unroll
    for (int ii = 0; ii < 8; ++ii)
      sH[(mb + ii) * 128 + n0 + m] = acc[ii] + bv;
  }
  __syncthreads();

  // a[n] = <other[n], master> for n=1..7 : wave n reduces 128 products.
  if (wv >= 1) {
    float partial = 0.0f;
#pragma unroll
    for (int j = 0; j < 4; ++j) {
      const int e = lane * 4 + j;
      partial += sH[wv * 128 + e] * sH[e];
    }
#pragma unroll
    for (int off = 16; off >= 1; off >>= 1)
      partial += __shfl_xor(partial, off, 32);
    if (lane == 0) sA[wv] = partial;
  }
  __syncthreads();

  // agg[e] = sum_n a[n] * other[n][e]
  if (tid < 128) {
    float s = 0.0f;
#pragma unroll
    for (int n = 1; n < 8; ++n) s += sA[n] * sH[n * 128 + tid];
    sAgg[tid] = s;
  }
  __syncthreads();

  // t1 = relu(agg @ m1a_w + b)   (1x128 GEMV)
  if (tid < 128) {
    float s = m1a_b[tid];
    for (int kx = 0; kx < 128; ++kx) s += sAgg[kx] * m1a_w[kx * 128 + tid];
    sT1[tid] = s > 0.0f ? s : 0.0f;
  }
  __syncthreads();

  // q_merged = q1 + t1 @ m1b_w + b   (1x256 GEMV)
  {
    float s = m1b_b[tid];
    for (int kx = 0; kx < 128; ++kx) s += sT1[kx] * m1b_w[kx * 256 + tid];
    sQm[tid] = sQ[tid] + s;
  }
  __syncthreads();

  // t2 = relu(q_merged @ reg_w1 + b)
  {
    float s = reg_b1[tid];
    for (int kx = 0; kx < 256; ++kx) s += sQm[kx] * reg_w1[kx * 256 + tid];
    sT2[tid] = s > 0.0f ? s : 0.0f;
  }
  __syncthreads();

  // pred = t2 @ reg_w2 + b   (3 outputs)
  if (tid < 3) {
    float s = reg_b2[tid];
    for (int kx = 0; kx < 256; ++kx) s += sT2[kx] * reg_w2[kx * 3 + tid];
    out[bk * 3 + tid] = s;
  }
}

// ---------------------------------------------------------------------------
extern "C" void kernel_launch(void* const* d_in, const int* in_sizes, int n_in,
                              void* d_out, int out_size, void* d_ws, size_t ws_size,
                              hipStream_t stream) {
  const float* img_feat = (const float*)d_in[0];
  const float* cam_intr = (const float*)d_in[1];
  const float* cam_extr = (const float*)d_in[2];
  const float* center   = (const float*)d_in[3];
  const float* pe_w1    = (const float*)d_in[4];
  const float* pe_b1    = (const float*)d_in[5];
  const float* pe_w2    = (const float*)d_in[6];
  const float* pe_b2    = (const float*)d_in[7];
  const float* proj_w   = (const float*)d_in[8];
  const float* proj_b   = (const float*)d_in[9];
  const float* m0a_w    = (const float*)d_in[10];
  const float* m0a_b    = (const float*)d_in[11];
  const float* m0b_w    = (const float*)d_in[12];
  const float* m0b_b    = (const float*)d_in[13];
  const float* m1a_w    = (const float*)d_in[14];
  const float* m1a_b    = (const float*)d_in[15];
  const float* m1b_w    = (const float*)d_in[16];
  const float* m1b_b    = (const float*)d_in[17];
  const float* reg_w1   = (const float*)d_in[18];
  const float* reg_b1   = (const float*)d_in[19];
  const float* reg_w2   = (const float*)d_in[20];
  const float* reg_b2   = (const float*)d_in[21];
  float* out = (float*)d_out;

  char* ws = (char*)d_ws;
  int*   ws_idx  = (int*)(ws);              // 128 ints
  float* ws_mask = (float*)(ws + 512);      // 128 floats
  float* ws_q    = (float*)(ws + 1024);     // 1024 x 256 floats (1 MB)

  k_ball<<<4, 256, 0, stream>>>(cam_intr, cam_extr, center, ws_idx, ws_mask);
  k_qproj<<<64, 256, 0, stream>>>(img_feat, cam_intr, cam_extr,
                                  pe_w1, pe_b1, pe_w2, pe_b2, proj_w, proj_b,
                                  ws_idx, ws_mask, ws_q);
  k_head<<<128, 256, 0, stream>>>(ws_q, m0a_w, m0a_b, m0b_w, m0b_b,
                                  m1a_w, m1a_b, m1b_w, m1b_b,
                                  reg_w1, reg_b1, reg_w2, reg_b2, out);
}